// Bayesian_83837761618203
// MI455X (gfx1250) — compile-verified
//
#include <hip/hip_runtime.h>
#include <hip/hip_bf16.h>
#include <math.h>

#define BATCH 512
#define IN_DIM 1024
#define OUT_DIM 1024

typedef float v2f __attribute__((ext_vector_type(2)));
typedef float v4f __attribute__((ext_vector_type(4)));
typedef float v8f __attribute__((ext_vector_type(8)));

// ---------------------------------------------------------------------------
// Kernel 1: sp[i,o] = softplus(log_std[i,o])  (4 MB, stays L2-resident)
// ---------------------------------------------------------------------------
__global__ __launch_bounds__(256) void softplus_kernel(
    const float* __restrict__ log_std, float* __restrict__ sp, int n) {
  int idx = blockIdx.x * 256 + threadIdx.x;
  if (idx < n) {
    float z = log_std[idx];
    // log_std is tiny (|z| < 0.06) so no overflow guard needed, but keep one.
    sp[idx] = (z > 20.0f) ? z : log1pf(expf(z));
  }
}

// ---------------------------------------------------------------------------
// Kernel 2: out = x @ mean + bias   via V_WMMA_F32_16X16X4_F32 (fp32 WMMA)
// One 16x16 output tile per wave32.  2048 tiles total -> 256 blocks x 8 waves.
//
// A (16x4 f32) lane layout: lanes 0-15 hold M=lane, {K=k0,K=k0+1} in {v0,v1};
// lanes 16-31 hold M=lane-16, {K=k0+2,K=k0+3}.
// B (4x16 f32): lanes 0-15 N=lane hold rows {K=k0,k0+1}; lanes 16-31 rows
// {K=k0+2,k0+3} (mirrors C/D striping).
// C/D (16x16 f32, 8 VGPRs): vgpr r -> M = r + (lane>=16 ? 8 : 0), N = lane&15.
// ---------------------------------------------------------------------------
__global__ __launch_bounds__(256) void gemm_mean_wmma(
    const float* __restrict__ x, const float* __restrict__ mean,
    const float* __restrict__ bias, float* __restrict__ out) {
  const int wave = threadIdx.x >> 5;                  // 0..7
  const int lane = threadIdx.x & 31;
  const int tile = blockIdx.x * 8 + wave;             // 0..2047
  const int tm = tile / (OUT_DIM / 16);               // 0..31 (batch tiles)
  const int tn = tile % (OUT_DIM / 16);               // 0..63 (out tiles)
  const int row0 = tm * 16;
  const int col0 = tn * 16;

  const int m    = lane & 15;
  const int n    = lane & 15;
  const int koff = (lane >> 4) * 2;                   // 0 or 2

  v8f acc = {};
  const float* xrow = x + (size_t)(row0 + m) * IN_DIM;

  #pragma unroll 4
  for (int k = 0; k < IN_DIM; k += 4) {
    // A fragment: two consecutive K's -> single b64 load
    v2f a = *(const v2f*)(xrow + k + koff);
    // B fragment: rows (k+koff) and (k+koff+1) of mean, column col0+n
    v2f b;
    b.x = mean[(size_t)(k + koff)     * OUT_DIM + col0 + n];
    b.y = mean[(size_t)(k + koff + 1) * OUT_DIM + col0 + n];
    acc = __builtin_amdgcn_wmma_f32_16x16x4_f32(
        /*neg_a=*/false, a, /*neg_b=*/false, b,
        /*c_mod=*/(short)0, acc, /*reuse_a=*/false, /*reuse_b=*/false);
  }

  const int mhalf = (lane >> 4) * 8;
  const float bv = bias[col0 + n];
  #pragma unroll
  for (int r = 0; r < 8; ++r) {
    out[(size_t)(row0 + r + mhalf) * OUT_DIM + col0 + n] = acc[r] + bv;
  }
}

// ---------------------------------------------------------------------------
// Kernel 3 (dominant, HBM-bound): out[b,o] += sum_i x[b,i]*sp[i,o]*W[b,i,o]
// Streams the 2 GiB W tensor exactly once with non-temporal b128 loads.
// grid = (512 batches, 2 column tiles), block = 128 threads (4 waves),
// each thread owns one float4 of the output row.
// ---------------------------------------------------------------------------
__global__ __launch_bounds__(128) void stream_w_kernel(
    const float* __restrict__ W, const float* __restrict__ x,
    const float* __restrict__ sp, float* __restrict__ out) {
  __shared__ float xrow[IN_DIM];
  const int b  = blockIdx.x;                          // 0..511
  const int o4 = blockIdx.y * 128 + threadIdx.x;      // float4 index, 0..255

  // Stage this batch's x row (4 KB) into LDS; inner loop reads are broadcasts.
  for (int i = threadIdx.x; i < IN_DIM; i += 128) {
    xrow[i] = x[(size_t)b * IN_DIM + i];
  }
  __syncthreads();

  const v4f* __restrict__ Wp = (const v4f*)(W + (size_t)b * IN_DIM * OUT_DIM);
  const v4f* __restrict__ Sp = (const v4f*)sp;

  v4f acc = {0.f, 0.f, 0.f, 0.f};
  #pragma unroll 4
  for (int i = 0; i < IN_DIM; ++i) {
    const float xv = xrow[i];                                   // ds broadcast
    const v4f s = Sp[(size_t)i * (OUT_DIM / 4) + o4];           // L2 hit
    const v4f w = __builtin_nontemporal_load(
        &Wp[(size_t)i * (OUT_DIM / 4) + o4]);                   // stream HBM
    acc += (s * xv) * w;                                        // 2 VALU/elt
  }

  float* op = out + (size_t)b * OUT_DIM + (size_t)o4 * 4;
  v4f prev = *(const v4f*)op;                                   // x@mean+bias
  *(v4f*)op = prev + acc;
}

// ---------------------------------------------------------------------------
extern "C" void kernel_launch(void* const* d_in, const int* in_sizes, int n_in,
                              void* d_out, int out_size, void* d_ws, size_t ws_size,
                              hipStream_t stream) {
  const float* x       = (const float*)d_in[0];  // [512,1024]
  const float* W       = (const float*)d_in[1];  // [512,1024,1024]
  const float* mean    = (const float*)d_in[2];  // [1024,1024]
  const float* log_std = (const float*)d_in[3];  // [1024,1024]
  const float* bias    = (const float*)d_in[4];  // [1024]
  float* out = (float*)d_out;                    // [512,1024]
  float* sp  = (float*)d_ws;                     // [1024,1024] scratch (4 MB)

  const int n_sp = IN_DIM * OUT_DIM;

  // 1) sp = softplus(log_std)
  softplus_kernel<<<(n_sp + 255) / 256, 256, 0, stream>>>(log_std, sp, n_sp);

  // 2) out = x @ mean + bias  (WMMA fp32)
  gemm_mean_wmma<<<(BATCH / 16) * (OUT_DIM / 16) / 8, 256, 0, stream>>>(
      x, mean, bias, out);

  // 3) out += sum_i x[b,i] * sp[i,o] * W[b,i,o]  (streams W once)
  stream_w_kernel<<<dim3(BATCH, OUT_DIM / 512), 128, 0, stream>>>(
      W, x, sp, out);
}